// GNNActor_27195732918295
// MI455X (gfx1250) — compile-verified
//
#include <hip/hip_runtime.h>
#include <math.h>

typedef float v2f __attribute__((ext_vector_type(2)));
typedef float v8f __attribute__((ext_vector_type(8)));

#define C_IN 64
#define HID  32

// ---------------- degree / normalization helpers ----------------

__global__ void init_kernel(float* deg, float* dsum, int N) {
    int i = blockIdx.x * blockDim.x + threadIdx.x;
    if (i < N) deg[i] = 1.0f;               // self-loop
    if (i == 0) *dsum = 0.0f;
}

__global__ void degree_kernel(const int* __restrict__ col, float* deg, int E) {
    int e = blockIdx.x * blockDim.x + threadIdx.x;
    if (e < E) atomicAdd(&deg[col[e]], 1.0f);
}

__global__ void rsqrt_kernel(float* deg, int N) {
    int i = blockIdx.x * blockDim.x + threadIdx.x;
    if (i < N) deg[i] = rsqrtf(fmaxf(deg[i], 1e-12f));
}

// ---------------- XW = state @ W_gcn via fp32 WMMA 16x16x4 ----------------
// A layout (32-bit A 16x4): lanes 0-15 hold K={0,1} in VGPR{0,1}; lanes 16-31 hold K={2,3}.
// B layout (row striped across lanes per VGPR): VGPR v = row K=v (lanes 0-15), K=v+2 (lanes 16-31).
// C/D layout: VGPR r = row r (lanes 0-15), row r+8 (lanes 16-31).

__global__ void gemm_wmma_kernel(const float* __restrict__ A,
                                 const float* __restrict__ W,
                                 float* __restrict__ D, int N) {
    const int wavesPerBlock = blockDim.x >> 5;
    int wave  = blockIdx.x * wavesPerBlock + (threadIdx.x >> 5);
    int lane  = threadIdx.x & 31;
    int half  = lane >> 4;
    int l     = lane & 15;
    const int tilesN = C_IN / 16;
    int tileM = wave / tilesN;
    int tileN = wave % tilesN;
    if (tileM * 16 >= N) return;    // wave-uniform -> EXEC all-ones for WMMA

    int m = tileM * 16 + l;
    int n = tileN * 16 + l;

    v8f c = {0.f, 0.f, 0.f, 0.f, 0.f, 0.f, 0.f, 0.f};
    const float* arow = A + (size_t)m * C_IN;

#pragma unroll
    for (int k0 = 0; k0 < C_IN; k0 += 4) {
        int ka = k0 + 2 * half;
        v2f a, b;
        a.x = arow[ka];
        a.y = arow[ka + 1];
        b.x = W[(size_t)ka * C_IN + n];
        b.y = W[(size_t)(ka + 1) * C_IN + n];
        c = __builtin_amdgcn_wmma_f32_16x16x4_f32(false, a, false, b,
                                                  (short)0, c, false, false);
    }

#pragma unroll
    for (int r = 0; r < 8; ++r) {
        int row = tileM * 16 + r + 8 * half;
        D[(size_t)row * C_IN + n] = c[r];
    }
}

// ---------------- self-loop term: acc = dinv(i)^2 * xw(i) ----------------

__global__ void selfloop_kernel(const float* __restrict__ xw,
                                const float* __restrict__ dinv,
                                float* __restrict__ acc, long long total) {
    long long t = (long long)blockIdx.x * blockDim.x + threadIdx.x;
    if (t >= total) return;
    int i = (int)(t >> 6);
    float dv = dinv[i];
    acc[t] = dv * dv * xw[t];
}

// ---------------- edge scatter: acc[col] += dinv[row]*dinv[col]*xw[row] ----------------

__global__ void scatter_kernel(const float* __restrict__ xw,
                               const float* __restrict__ dinv,
                               const int* __restrict__ row,
                               const int* __restrict__ col,
                               float* acc, long long total) {
    long long t = (long long)blockIdx.x * blockDim.x + threadIdx.x;
    if (t >= total) return;
    int e  = (int)(t >> 6);
    int ch = (int)(t & 63);
    int r  = row[e];
    int cl = col[e];
    float norm = dinv[r] * dinv[cl];
    atomicAdd(&acc[(size_t)cl * C_IN + ch], norm * xw[(size_t)r * C_IN + ch]);
}

// ---------------- WMMA MLP: bias+ReLU+residual, L1, L2, L3, softplus, sum ----------------
// 4 waves/block, each wave owns a 16-node tile.
// LDS strides: sx rows padded to 68 floats (A reads bank (4l+ka)%64, conflict-free),
// sh rows padded to 36 floats (36l%64 hits 16 distinct banks; D writes & A reads clean).

#define XSTRIDE 68
#define HSTRIDE 36

__global__ void __launch_bounds__(128)
mlp_wmma_kernel(const float* __restrict__ acc,
                const float* __restrict__ state,
                const float* __restrict__ b_gcn,
                const float* __restrict__ W1, const float* __restrict__ b1,
                const float* __restrict__ W2, const float* __restrict__ b2,
                const float* __restrict__ W3, const float* __restrict__ b3,
                float* __restrict__ conc, float* dsum, int N) {
    __shared__ float sW1[C_IN * HID];       // 64x32
    __shared__ float sW2[HID * HID];        // 32x32
    __shared__ float sW3[HID];
    __shared__ float sb1[HID];
    __shared__ float sb2[HID];
    __shared__ float sbg[C_IN];
    __shared__ float sx[4 * 16 * XSTRIDE];  // per-wave x staging (16 rows x 64, pad 68)
    __shared__ float sh[4 * 16 * HSTRIDE];  // per-wave hidden staging (16 rows x 32, pad 36)
    __shared__ float red[128];

    const int tid    = threadIdx.x;
    const int waveId = tid >> 5;
    const int lane   = tid & 31;
    const int half   = lane >> 4;
    const int l      = lane & 15;

    for (int k = tid; k < C_IN * HID; k += blockDim.x) sW1[k] = W1[k];
    for (int k = tid; k < HID * HID; k += blockDim.x)  sW2[k] = W2[k];
    if (tid < HID) { sW3[tid] = W3[tid]; sb1[tid] = b1[tid]; sb2[tid] = b2[tid]; }
    if (tid >= HID && tid < HID + C_IN) sbg[tid - HID] = b_gcn[tid - HID];
    __syncthreads();

    const int m0 = (blockIdx.x * 4 + waveId) * 16;   // first node of this wave's tile
    float sp = 0.f;

    if (m0 < N) {                                    // wave-uniform guard
        float* sxw = sx + waveId * (16 * XSTRIDE);
        float* shw = sh + waveId * (16 * HSTRIDE);

        // ---- stage x = relu(acc + b_gcn) + state into LDS (coalesced float4) ----
#pragma unroll
        for (int it = 0; it < 8; ++it) {
            int f   = it * 128 + lane * 4;           // 0..1023
            int row = f >> 6;
            int k   = f & 63;
            float4 a = *(const float4*)(acc   + (size_t)(m0 + row) * C_IN + k);
            float4 s = *(const float4*)(state + (size_t)(m0 + row) * C_IN + k);
            float4 x;
            x.x = fmaxf(a.x + sbg[k + 0], 0.f) + s.x;
            x.y = fmaxf(a.y + sbg[k + 1], 0.f) + s.y;
            x.z = fmaxf(a.z + sbg[k + 2], 0.f) + s.z;
            x.w = fmaxf(a.w + sbg[k + 3], 0.f) + s.w;
            *(float4*)(sxw + row * XSTRIDE + k) = x;
        }

        // ---- layer 1: [16x64] @ [64x32] -> two 16x16 WMMA column tiles ----
        v8f d0 = {0.f,0.f,0.f,0.f,0.f,0.f,0.f,0.f};
        v8f d1 = {0.f,0.f,0.f,0.f,0.f,0.f,0.f,0.f};
#pragma unroll
        for (int k0 = 0; k0 < C_IN; k0 += 4) {
            int ka = k0 + 2 * half;
            v2f a, b0, b1v;
            a.x   = sxw[l * XSTRIDE + ka];
            a.y   = sxw[l * XSTRIDE + ka + 1];
            b0.x  = sW1[ka * HID + l];
            b0.y  = sW1[(ka + 1) * HID + l];
            b1v.x = sW1[ka * HID + l + 16];
            b1v.y = sW1[(ka + 1) * HID + l + 16];
            d0 = __builtin_amdgcn_wmma_f32_16x16x4_f32(false, a, false, b0,
                                                       (short)0, d0, false, false);
            d1 = __builtin_amdgcn_wmma_f32_16x16x4_f32(false, a, false, b1v,
                                                       (short)0, d1, false, false);
        }
        // bias + leaky relu, re-layout D -> A through LDS
#pragma unroll
        for (int r = 0; r < 8; ++r) {
            int rowm = r + 8 * half;
            float v0 = d0[r] + sb1[l];
            float v1 = d1[r] + sb1[l + 16];
            shw[rowm * HSTRIDE + l]      = v0 > 0.f ? v0 : 0.01f * v0;
            shw[rowm * HSTRIDE + l + 16] = v1 > 0.f ? v1 : 0.01f * v1;
        }

        // ---- layer 2: [16x32] @ [32x32] -> two 16x16 WMMA column tiles ----
        v8f e0 = {0.f,0.f,0.f,0.f,0.f,0.f,0.f,0.f};
        v8f e1 = {0.f,0.f,0.f,0.f,0.f,0.f,0.f,0.f};
#pragma unroll
        for (int k0 = 0; k0 < HID; k0 += 4) {
            int ka = k0 + 2 * half;
            v2f a, b0, b1v;
            a.x   = shw[l * HSTRIDE + ka];
            a.y   = shw[l * HSTRIDE + ka + 1];
            b0.x  = sW2[ka * HID + l];
            b0.y  = sW2[(ka + 1) * HID + l];
            b1v.x = sW2[ka * HID + l + 16];
            b1v.y = sW2[(ka + 1) * HID + l + 16];
            e0 = __builtin_amdgcn_wmma_f32_16x16x4_f32(false, a, false, b0,
                                                       (short)0, e0, false, false);
            e1 = __builtin_amdgcn_wmma_f32_16x16x4_f32(false, a, false, b1v,
                                                       (short)0, e1, false, false);
        }
        // bias + leaky relu, write h2 back into the same per-wave buffer
#pragma unroll
        for (int r = 0; r < 8; ++r) {
            int rowm = r + 8 * half;
            float v0 = e0[r] + sb2[l];
            float v1 = e1[r] + sb2[l + 16];
            shw[rowm * HSTRIDE + l]      = v0 > 0.f ? v0 : 0.01f * v0;
            shw[rowm * HSTRIDE + l + 16] = v1 > 0.f ? v1 : 0.01f * v1;
        }

        // ---- layer 3: per-node dot(h2, W3); halves split across lane pairs ----
        float part = 0.f;
#pragma unroll
        for (int n = 0; n < 16; ++n) {
            int nn = half * 16 + n;
            part = fmaf(shw[l * HSTRIDE + nn], sW3[nn], part);
        }
        part += __shfl_xor(part, 16, 32);            // combine the two halves
        float s3 = part + b3[0];
        // stable softplus
        float spv = fmaxf(s3, 0.f) + log1pf(expf(-fabsf(s3)));
        if (half == 0) {
            conc[m0 + l] = spv;
            sp = spv;
        }
    }

    red[tid] = sp;
    __syncthreads();
    for (int s = blockDim.x >> 1; s > 0; s >>= 1) {
        if (tid < s) red[tid] += red[tid + s];
        __syncthreads();
    }
    if (tid == 0) atomicAdd(dsum, red[0]);
}

__global__ void normalize_kernel(const float* __restrict__ conc,
                                 const float* __restrict__ dsum,
                                 float* __restrict__ out, int N) {
    int i = blockIdx.x * blockDim.x + threadIdx.x;
    if (i < N) out[i] = conc[i] / (*dsum + 1e-20f);
}

// ---------------- launcher ----------------

extern "C" void kernel_launch(void* const* d_in, const int* in_sizes, int n_in,
                              void* d_out, int out_size, void* d_ws, size_t ws_size,
                              hipStream_t stream) {
    const float* state = (const float*)d_in[0];
    const int*   edge  = (const int*)  d_in[1];
    const float* W_gcn = (const float*)d_in[2];
    const float* b_gcn = (const float*)d_in[3];
    const float* W1    = (const float*)d_in[4];
    const float* b1    = (const float*)d_in[5];
    const float* W2    = (const float*)d_in[6];
    const float* b2    = (const float*)d_in[7];
    const float* W3    = (const float*)d_in[8];
    const float* b3    = (const float*)d_in[9];

    const int N = in_sizes[0] / C_IN;
    const int E = in_sizes[1] / 2;
    const int* row = edge;        // sources
    const int* col = edge + E;    // targets

    // workspace partition (~51 MB)
    float* xw   = (float*)d_ws;                 // N*64
    float* acc  = xw   + (size_t)N * C_IN;      // N*64
    float* dinv = acc  + (size_t)N * C_IN;      // N
    float* conc = dinv + N;                     // N
    float* dsum = conc + N;                     // 1

    const int B = 256;

    init_kernel  <<<(N + B - 1) / B, B, 0, stream>>>(dinv, dsum, N);
    degree_kernel<<<(E + B - 1) / B, B, 0, stream>>>(col, dinv, E);
    rsqrt_kernel <<<(N + B - 1) / B, B, 0, stream>>>(dinv, N);

    int waves = (N / 16) * (C_IN / 16);
    gemm_wmma_kernel<<<(waves + 7) / 8, 256, 0, stream>>>(state, W_gcn, xw, N);

    long long nc = (long long)N * C_IN;
    selfloop_kernel<<<(int)((nc + B - 1) / B), B, 0, stream>>>(xw, dinv, acc, nc);

    long long ec = (long long)E * C_IN;
    scatter_kernel<<<(int)((ec + B - 1) / B), B, 0, stream>>>(xw, dinv, row, col, acc, ec);

    // 4 waves/block, 16 nodes per wave -> 64 nodes per block
    mlp_wmma_kernel<<<(N + 63) / 64, 128, 0, stream>>>(acc, state, b_gcn, W1, b1,
                                                       W2, b2, W3, b3, conc, dsum, N);

    normalize_kernel<<<(N + B - 1) / B, B, 0, stream>>>(conc, dsum, (float*)d_out, N);
}